// ChannelAdder_41102837023348
// MI455X (gfx1250) — compile-verified
//
#include <hip/hip_runtime.h>
#include <math.h>

#define TX 32
#define TY 8
#define R2X (TX + 4)          // halo-2 region width  = 36
#define R2Y (TY + 4)          // halo-2 region height = 12
#define A2  (R2X * R2Y)       // 432
#define R1X (TX + 2)          // halo-1 region width  = 34
#define R1Y (TY + 2)          // halo-1 region height = 10
#define A1  (R1X * R1Y)       // 340

#define NB   16
#define CI   3
#define HH   512
#define WW   512
#define HWSZ (HH * WW)
#define COUT 20

// CDNA5 async global->LDS path (ASYNCcnt). Guarded so compile never fails if
// the toolchain lacks the builtins; fallback is load->ds_store.
#if defined(__HIP_DEVICE_COMPILE__) && defined(__gfx1250__) && \
    __has_builtin(__builtin_amdgcn_global_load_async_to_lds_b32) && \
    __has_builtin(__builtin_amdgcn_s_wait_asynccnt)
#define USE_ASYNC_LDS 1
#define AS1CAST(p) ((__attribute__((address_space(1))) int*)(p))
#define AS3CAST(p) ((__attribute__((address_space(3))) int*)(p))
#else
#define USE_ASYNC_LDS 0
#endif

__global__ __launch_bounds__(TX * TY) void sobel_fused_kernel(
    const float* __restrict__ x, float* __restrict__ out)
{
    __shared__ float sIN[CI * A2];   // input, halo 2
    __shared__ float sEN[2 * A2];    // e_norm, halo 2 (0 outside image)
    __shared__ float sS1[10 * A1];   // stage-1 channels, halo 1 (0 outside image)

    const int tx  = threadIdx.x;
    const int ty  = threadIdx.y;
    const int tid = ty * TX + tx;
    const int w0  = blockIdx.x * TX;
    const int h0  = blockIdx.y * TY;
    const int n   = blockIdx.z;

    const float* xn = x + (size_t)n * CI * HWSZ;

    // ---- Stage 1: stage input tile (3 ch, halo 2) into LDS -----------------
    for (int i = tid; i < CI * A2; i += TX * TY) {
        int c  = i / A2;
        int r  = i - c * A2;
        int ly = r / R2X;
        int lx = r - ly * R2X;
        int gh = h0 - 2 + ly;
        int gw = w0 - 2 + lx;
        bool inb = (gh >= 0) && (gh < HH) && (gw >= 0) && (gw < WW);
#if USE_ASYNC_LDS
        if (inb) {
            const float* src = xn + (size_t)c * HWSZ + (size_t)gh * WW + gw;
            __builtin_amdgcn_global_load_async_to_lds_b32(
                AS1CAST(src), AS3CAST(&sIN[i]), 0, 0);
        } else {
            sIN[i] = 0.0f;
        }
#else
        sIN[i] = inb ? xn[(size_t)c * HWSZ + (size_t)gh * WW + gw] : 0.0f;
#endif
    }
#if USE_ASYNC_LDS
    __builtin_amdgcn_s_wait_asynccnt(0);
#endif
    __syncthreads();

    // ---- Stage 2: e_norm on halo-2 region (zero outside the image) ---------
    for (int i = tid; i < A2; i += TX * TY) {
        int ly = i / R2X;
        int lx = i - ly * R2X;
        int gh = h0 - 2 + ly;
        int gw = w0 - 2 + lx;
        float e0 = sIN[i];
        float e1 = sIN[A2 + i];
        float en0 = 0.0f, en1 = 0.0f;
        if ((gh >= 0) && (gh < HH) && (gw >= 0) && (gw < WW)) {
            float m = sqrtf(e0 * e0 + e1 * e1);
            en0 = e0 / m;
            en1 = e1 / m;
        }
        sEN[i]      = en0;
        sEN[A2 + i] = en1;
    }
    __syncthreads();

    // ---- Stage 3: 10 stage-1 channels on halo-1 region ---------------------
    for (int i = tid; i < A1; i += TX * TY) {
        int ly = i / R1X;
        int lx = i - ly * R1X;
        int gh = h0 - 1 + ly;
        int gw = w0 - 1 + lx;
        int j  = (ly + 1) * R2X + (lx + 1);   // same point in halo-2 coords
        float c0 = 0.f, c1 = 0.f, c2 = 0.f, c3 = 0.f, c4 = 0.f;
        float c5 = 0.f, c6 = 0.f, c7 = 0.f, c8 = 0.f, c9 = 0.f;
        if ((gh >= 0) && (gh < HH) && (gw >= 0) && (gw < WW)) {
            float e0  = sIN[j];
            float e1  = sIN[A2 + j];
            float dlt = sIN[2 * A2 + j];
            float mag = sqrtf(e0 * e0 + e1 * e1);
            float ang = 0.5f * atan2f(e0, e1);
            float en0 = sEN[j];
            float en1 = sEN[A2 + j];
            // 4-neighbor cosine sims; out-of-image neighbors give exact 0
            float du = en0 * sEN[j + R2X] + en1 * sEN[A2 + j + R2X];
            float dd = en0 * sEN[j - R2X] + en1 * sEN[A2 + j - R2X];
            float dl = en0 * sEN[j + 1]   + en1 * sEN[A2 + j + 1];
            float dr = en0 * sEN[j - 1]   + en1 * sEN[A2 + j - 1];
            c0 = e0; c1 = e1; c2 = mag; c3 = ang; c4 = en0; c5 = en1;
            c6 = e0 * e1; c7 = dlt;
            c8 = fmaxf(fmaxf(du, dd), fmaxf(dl, dr));
            c9 = fminf(fminf(du, dd), fminf(dl, dr));
        }
        sS1[0 * A1 + i] = c0; sS1[1 * A1 + i] = c1;
        sS1[2 * A1 + i] = c2; sS1[3 * A1 + i] = c3;
        sS1[4 * A1 + i] = c4; sS1[5 * A1 + i] = c5;
        sS1[6 * A1 + i] = c6; sS1[7 * A1 + i] = c7;
        sS1[8 * A1 + i] = c8; sS1[9 * A1 + i] = c9;
    }
    __syncthreads();

    // ---- Stage 4: emit 20 channels (NT stores; output >> L2 capacity) ------
    const int gh = h0 + ty;
    const int gw = w0 + tx;
    float* on = out + (size_t)n * COUT * HWSZ + (size_t)gh * WW + gw;
    const int j1 = (ty + 1) * R1X + (tx + 1);
#pragma unroll
    for (int c = 0; c < 10; ++c) {
        const float* S = &sS1[c * A1];
        float pC  = S[j1];
        float pA  = S[j1 - R1X - 1];
        float pB  = S[j1 - R1X];
        float pCc = S[j1 - R1X + 1];
        float pD  = S[j1 - 1];
        float pF  = S[j1 + 1];
        float pG  = S[j1 + R1X - 1];
        float pH  = S[j1 + R1X];
        float pI  = S[j1 + R1X + 1];
        float gx = (pCc - pA) + 2.0f * (pF - pD) + (pI - pG);
        float gy = (pG - pA) + 2.0f * (pH - pB) + (pI - pCc);
        __builtin_nontemporal_store(pC, on + (size_t)c * HWSZ);
        __builtin_nontemporal_store(sqrtf(gx * gx + gy * gy),
                                    on + (size_t)(10 + c) * HWSZ);
    }
}

extern "C" void kernel_launch(void* const* d_in, const int* in_sizes, int n_in,
                              void* d_out, int out_size, void* d_ws, size_t ws_size,
                              hipStream_t stream) {
    (void)in_sizes; (void)n_in; (void)out_size; (void)d_ws; (void)ws_size;
    const float* x  = (const float*)d_in[0];
    float* out      = (float*)d_out;
    dim3 grid(WW / TX, HH / TY, NB);
    dim3 block(TX, TY, 1);
    hipLaunchKernelGGL(sobel_fused_kernel, grid, block, 0, stream, x, out);
}